// GraphSparseConvolution_37941741093301
// MI455X (gfx1250) — compile-verified
//
#include <hip/hip_runtime.h>
#include <stdint.h>

#define OUT_F 64
#define KEEP_PROB 0.9f
#define INV_KEEP (1.0f / 0.9f)

typedef __attribute__((ext_vector_type(2))) float v2f;
typedef __attribute__((ext_vector_type(8))) float v8f;

// ---------------- threefry2x32 (JAX-compatible, 20 rounds) ----------------
__device__ __forceinline__ uint32_t rotl32(uint32_t x, int n) {
  return (x << n) | (x >> (32 - n));
}

__device__ __forceinline__ void threefry2x32(uint32_t k0, uint32_t k1,
                                             uint32_t x0, uint32_t x1,
                                             uint32_t& o0, uint32_t& o1) {
  const uint32_t ks2 = k0 ^ k1 ^ 0x1BD11BDAu;
  x0 += k0; x1 += k1;
#define TF4A { x0+=x1; x1=rotl32(x1,13); x1^=x0; x0+=x1; x1=rotl32(x1,15); x1^=x0; \
               x0+=x1; x1=rotl32(x1,26); x1^=x0; x0+=x1; x1=rotl32(x1, 6); x1^=x0; }
#define TF4B { x0+=x1; x1=rotl32(x1,17); x1^=x0; x0+=x1; x1=rotl32(x1,29); x1^=x0; \
               x0+=x1; x1=rotl32(x1,16); x1^=x0; x0+=x1; x1=rotl32(x1,24); x1^=x0; }
  TF4A; x0 += k1;  x1 += ks2 + 1u;
  TF4B; x0 += ks2; x1 += k0  + 2u;
  TF4A; x0 += k0;  x1 += k1  + 3u;
  TF4B; x0 += k1;  x1 += ks2 + 4u;
  TF4A; x0 += ks2; x1 += k0  + 5u;
#undef TF4A
#undef TF4B
  o0 = x0; o1 = x1;
}

// dropout scale for nnz index i: floor(0.9 + U[0,1)) / 0.9, U from key(42)
__device__ __forceinline__ float dropout_scale(int i, int nnz) {
  const uint32_t K0 = 0u, K1 = 42u;  // jax.random.key(42) -> (hi, lo)
  const int H = (nnz + 1) >> 1;      // split-iota halves
  uint32_t o0, o1, bits;
  if (i < H) { threefry2x32(K0, K1, (uint32_t)i, (uint32_t)(i + H), o0, o1); bits = o0; }
  else       { threefry2x32(K0, K1, (uint32_t)(i - H), (uint32_t)i, o0, o1); bits = o1; }
  float u = __uint_as_float((bits >> 9) | 0x3F800000u) - 1.0f;  // [0,1)
  return floorf(KEEP_PROB + u) * INV_KEEP;
}

// ---------------- elementwise helpers ----------------
__global__ void zero4_kernel(float4* __restrict__ p, long n4) {
  long i = (long)blockIdx.x * blockDim.x + threadIdx.x;
  if (i < n4) p[i] = make_float4(0.f, 0.f, 0.f, 0.f);
}

__global__ void relu_kernel(float* __restrict__ p, long n) {
  long i = (long)blockIdx.x * blockDim.x + threadIdx.x;
  if (i < n) { float v = p[i]; p[i] = v > 0.f ? v : 0.f; }
}

// ---------------- densify sparse X (with dropout) into [N x F] ----------------
__global__ void densify_kernel(const float* __restrict__ xv, const int* __restrict__ xr,
                               const int* __restrict__ xc, float* __restrict__ Xd,
                               int nnz, int F) {
  int i = blockIdx.x * blockDim.x + threadIdx.x;
  if (i >= nnz) return;
  float v = xv[i] * dropout_scale(i, nnz);
  atomicAdd(&Xd[(size_t)xr[i] * F + xc[i]], v);
}

// ---------------- XW = Xdense @ W via V_WMMA_F32_16X16X4_F32 ----------------
// Block: 128 threads (4 waves). Each block: one 16-row M-tile, full N=64.
// Wave w owns N-tile [16w, 16w+16). A tile staged in LDS (516-float row pitch
// -> ds_load_b64 fragment reads land on distinct banks).
__global__ void __launch_bounds__(128)
xw_wmma_kernel(const float* __restrict__ Xd, const float* __restrict__ W,
               float* __restrict__ XW, int n_nodes, int F) {
  __shared__ __align__(16) float As[16][516];
  const int tid  = threadIdx.x;
  const int wave = tid >> 5;
  const int lane = tid & 31;
  const int m0   = blockIdx.x * 16;

  // Stage 16 x F tile of Xd, coalesced float4 loads.
  const int row4 = F >> 2;
  for (int idx = tid; idx < 16 * row4; idx += 128) {
    int r  = idx / row4;
    int c4 = idx - r * row4;
    float4 v = make_float4(0.f, 0.f, 0.f, 0.f);
    int gr = m0 + r;
    if (gr < n_nodes) v = ((const float4*)(Xd + (size_t)gr * F))[c4];
    *(float4*)&As[r][c4 * 4] = v;
  }
  __syncthreads();

  const int half = lane >> 4;   // 0: K0/K1, 1: K2/K3 (ISA A-matrix 16x4 layout)
  const int l16  = lane & 15;
  const int n0   = wave * 16;

  v8f acc = {};
#pragma unroll 4
  for (int k0 = 0; k0 < F; k0 += 4) {
    const int ks = k0 + 2 * half;
    // A fragment: row M=l16, (K=ks, K=ks+1) per lane
    float2 af = *(const float2*)&As[l16][ks];
    v2f a; a.x = af.x; a.y = af.y;
    // B fragment: VGPR0 holds row K=ks (N=n0+l16), VGPR1 row K=ks+1
    const float* wp = W + (size_t)ks * OUT_F + n0 + l16;
    v2f b; b.x = wp[0]; b.y = wp[OUT_F];
    acc = __builtin_amdgcn_wmma_f32_16x16x4_f32(false, a, false, b,
                                                (short)0, acc, false, false);
  }

  // C layout: VGPR r -> M=r (lanes 0-15) / M=r+8 (lanes 16-31), N = n0+l16
#pragma unroll
  for (int r = 0; r < 8; ++r) {
    int gm = m0 + r + 8 * half;
    if (gm < n_nodes) XW[(size_t)gm * OUT_F + n0 + l16] = acc[r];
  }
}

// ---------------- fallback: direct sparse scatter X@W (no dense ws) ----------------
__global__ void xw_scatter_kernel(const float* __restrict__ xv, const int* __restrict__ xr,
                                  const int* __restrict__ xc, const float* __restrict__ W,
                                  float* __restrict__ XW, int nnz) {
  int idx  = blockIdx.x * blockDim.x + threadIdx.x;
  int g    = idx >> 5;          // one wave32 group per nnz
  int lane = idx & 31;          // 2 outputs per lane
  if (g >= nnz) return;
  float v = xv[g] * dropout_scale(g, nnz);
  int r = xr[g], c = xc[g];
  float2 w = ((const float2*)(W + (size_t)c * OUT_F))[lane];
  atomicAdd(&XW[(size_t)r * OUT_F + 2 * lane + 0], v * w.x);
  atomicAdd(&XW[(size_t)r * OUT_F + 2 * lane + 1], v * w.y);
}

// ---------------- out += A @ XW (edge gather + scatter-atomic, L2-resident) --------
__global__ void spmm_edges_kernel(const float* __restrict__ adj_vals,
                                  const int* __restrict__ adj_rows,
                                  const int* __restrict__ adj_cols,
                                  const float* __restrict__ XW,
                                  float* __restrict__ out, int n_edges) {
  int idx  = blockIdx.x * blockDim.x + threadIdx.x;
  int e    = idx >> 5;          // one wave32 group per edge
  int lane = idx & 31;          // 2 outputs per lane (float2 gather)
  if (e >= n_edges) return;
  float av = adj_vals[e];
  int r = adj_rows[e], c = adj_cols[e];
  float2 m = ((const float2*)(XW + (size_t)c * OUT_F))[lane];
  atomicAdd(&out[(size_t)r * OUT_F + 2 * lane + 0], av * m.x);
  atomicAdd(&out[(size_t)r * OUT_F + 2 * lane + 1], av * m.y);
}

// ---------------- host ----------------
extern "C" void kernel_launch(void* const* d_in, const int* in_sizes, int n_in,
                              void* d_out, int out_size, void* d_ws, size_t ws_size,
                              hipStream_t stream) {
  const float* x_vals   = (const float*)d_in[0];
  const int*   x_rows   = (const int*)d_in[1];
  const int*   x_cols   = (const int*)d_in[2];
  const float* adj_vals = (const float*)d_in[3];
  const int*   adj_rows = (const int*)d_in[4];
  const int*   adj_cols = (const int*)d_in[5];
  const float* W        = (const float*)d_in[6];

  const int nnz     = in_sizes[0];
  const int n_edges = in_sizes[3];
  const int F       = in_sizes[6] / OUT_F;      // 512
  const int n_nodes = out_size / OUT_F;         // 100000
  float* out = (float*)d_out;

  float* xw = (float*)d_ws;
  const size_t xw_bytes = (size_t)n_nodes * OUT_F * sizeof(float);
  float* Xd = (float*)((char*)d_ws + xw_bytes);
  const size_t xd_bytes = (size_t)n_nodes * F * sizeof(float);
  const bool dense_path = (ws_size >= xw_bytes + xd_bytes) && (F <= 2064) && ((F & 3) == 0);

  const int TB = 256;

  // zero XW and out (accumulated via atomics)
  {
    long n4 = (long)(xw_bytes / 16);
    zero4_kernel<<<(unsigned)((n4 + TB - 1) / TB), TB, 0, stream>>>((float4*)xw, n4);
    long o4 = (long)out_size / 4;
    zero4_kernel<<<(unsigned)((o4 + TB - 1) / TB), TB, 0, stream>>>((float4*)out, o4);
  }

  if (dense_path) {
    long n4 = (long)(xd_bytes / 16);
    zero4_kernel<<<(unsigned)((n4 + TB - 1) / TB), TB, 0, stream>>>((float4*)Xd, n4);
    densify_kernel<<<(nnz + TB - 1) / TB, TB, 0, stream>>>(x_vals, x_rows, x_cols, Xd, nnz, F);
    int mtiles = (n_nodes + 15) / 16;
    xw_wmma_kernel<<<mtiles, 128, 0, stream>>>(Xd, W, xw, n_nodes, F);
  } else {
    long threads = (long)nnz * 32;
    xw_scatter_kernel<<<(unsigned)((threads + TB - 1) / TB), TB, 0, stream>>>(
        x_vals, x_rows, x_cols, W, xw, nnz);
  }

  {
    long threads = (long)n_edges * 32;
    spmm_edges_kernel<<<(unsigned)((threads + TB - 1) / TB), TB, 0, stream>>>(
        adj_vals, adj_rows, adj_cols, xw, out, n_edges);
  }

  relu_kernel<<<(unsigned)(((long)out_size + TB - 1) / TB), TB, 0, stream>>>(out, (long)out_size);
}